// LightGCN_81810537054280
// MI455X (gfx1250) — compile-verified
//
#include <hip/hip_runtime.h>

#define EMB 64
#define WTILE 32          // edges per wave tile (double-buffered)
#define EPW 128           // edges per wave
#define NWAVES 8          // waves per block (256 threads, wave32)
#define EPB (EPW * NWAVES)
#define NLAYERS 3
#define LAYER_SCALE 0.25f // 1/(NLAYERS+1)

typedef float v4f __attribute__((ext_vector_type(4)));

// ---------------------------------------------------------------------------
// CDNA5 helpers
// ---------------------------------------------------------------------------

// Low 32 bits of a generic pointer to __shared__ data == LDS byte offset.
__device__ __forceinline__ unsigned lds_off(const void* p) {
  return (unsigned)(unsigned long long)p;
}

// AS3 (LDS) volatile pointers, built from the 32-bit LDS offset (inttoptr),
// so reads lower to ds_load_b32.
typedef __attribute__((address_space(3))) volatile int   as3_vint;
typedef __attribute__((address_space(3))) volatile float as3_vfloat;

__device__ __forceinline__ as3_vint* lds_iptr(const void* p) {
  return (as3_vint*)lds_off(p);
}
__device__ __forceinline__ as3_vfloat* lds_fptr(const void* p) {
  return (as3_vfloat*)lds_off(p);
}

// gfx1250 async DMA: each active lane copies 4 bytes from the global address in vaddr to the LDS address in vdst, ASYNCcnt-tracked.
// NT hint: read-once edge streams must not evict the L2-resident tables.
__device__ __forceinline__ void async_ld_b32(unsigned ldsoff, const void* gaddr) {
  asm volatile("global_load_async_to_lds_b32 %0, %1, off th:TH_LOAD_NT"
               :: "v"(ldsoff), "v"(gaddr)
               : "memory");
}

__device__ __forceinline__ void wait_asynccnt0() {
  asm volatile("s_wait_asynccnt 0" ::: "memory");
}
// <=3 outstanding: the previous tile's 3 DMAs have retired (in-order completion)
__device__ __forceinline__ void wait_asynccnt3() {
  asm volatile("s_wait_asynccnt 3" ::: "memory");
}

// Native non-returning f32 atomic adds at device scope (RMW at L2).
__device__ __forceinline__ void atomic4_add_f32(float* dp, float x, float y,
                                                float z, float w) {
  asm volatile("global_atomic_add_f32 %0, %1, off scope:SCOPE_DEV"
               :: "v"(dp), "v"(x) : "memory");
  asm volatile("global_atomic_add_f32 %0, %1, off offset:4 scope:SCOPE_DEV"
               :: "v"(dp), "v"(y) : "memory");
  asm volatile("global_atomic_add_f32 %0, %1, off offset:8 scope:SCOPE_DEV"
               :: "v"(dp), "v"(z) : "memory");
  asm volatile("global_atomic_add_f32 %0, %1, off offset:12 scope:SCOPE_DEV"
               :: "v"(dp), "v"(w) : "memory");
}

// ---------------------------------------------------------------------------
// dst[sidx[e]] += vals[e] * src[gidx[e]]   (rows of EMB floats, atomic scatter)
// Wave-private double-buffered async staging of the edge streams; 16 lanes per
// edge (2 edges in flight per wave32), float4 per lane. No block barriers.
// ---------------------------------------------------------------------------
__global__ void __launch_bounds__(256)
scatter_spmm(float* __restrict__ dst, const float* __restrict__ src,
             const int* __restrict__ gidx, const int* __restrict__ sidx,
             const float* __restrict__ vals, int nEdges) {
  // per-wave: two buffers of WTILE entries per stream
  __shared__ __align__(16) int   sG[NWAVES][2][WTILE];
  __shared__ __align__(16) int   sS[NWAVES][2][WTILE];
  __shared__ __align__(16) float sW[NWAVES][2][WTILE];

  const int tid    = threadIdx.x;
  const int wid    = tid >> 5;       // wave id 0..7
  const int ln     = tid & 31;       // lane in wave
  const int group  = ln >> 4;        // 0..1 : which of the 2 edges in flight
  const int lane16 = ln & 15;        // which float4 of the 64-float row

  const int wStart = blockIdx.x * EPB + wid * EPW;
  if (wStart >= nEdges) return;      // barrier-free kernel: early exit is safe
  int wEnd = wStart + EPW;
  if (wEnd > nEdges) wEnd = nEdges;

  as3_vint*   vG = lds_iptr(&sG[wid][0][0]);
  as3_vint*   vS = lds_iptr(&sS[wid][0][0]);
  as3_vfloat* vW = lds_fptr(&sW[wid][0][0]);

  // stage one wave-tile (<=WTILE edges at e0) into buffer b : 3 async DMAs
  auto stage = [&](int b, int e0) {
    int cnt = wEnd - e0;
    if (cnt > WTILE) cnt = WTILE;
    if (ln < cnt) {
      async_ld_b32(lds_off(&sG[wid][b][ln]), gidx + e0 + ln);
      async_ld_b32(lds_off(&sS[wid][b][ln]), sidx + e0 + ln);
      async_ld_b32(lds_off(&sW[wid][b][ln]), vals + e0 + ln);
    }
  };

  int buf = 0;
  stage(0, wStart);

  for (int t0 = wStart; t0 < wEnd; t0 += WTILE) {
    const int nxt = t0 + WTILE;
    if (nxt < wEnd) {
      stage(buf ^ 1, nxt);   // issue next tile's DMA first...
      wait_asynccnt3();      // ...then wait only for the current tile's 3 ops
    } else {
      wait_asynccnt0();
    }

    // warm the streams two tiles ahead (HBM -> L2/WGP$)
    const int pf = t0 + 2 * WTILE + ln;
    if (pf < nEdges) {
      __builtin_prefetch(gidx + pf, 0, 0);
      __builtin_prefetch(sidx + pf, 0, 0);
      __builtin_prefetch(vals + pf, 0, 0);
    }

    int cnt = wEnd - t0;
    if (cnt > WTILE) cnt = WTILE;
    const int base = buf * WTILE;

    // 2 edges per iteration (one per 16-lane group)
    #pragma unroll 4
    for (int ii = 0; ii < WTILE / 2; ++ii) {
      const int e = ii * 2 + group;
      if (e < cnt) {
        const int   g = vG[base + e];
        const int   s = vS[base + e];
        const float w = vW[base + e];
        const float4* sp = reinterpret_cast<const float4*>(src + (size_t)g * EMB);
        const float4  v  = sp[lane16];
        float* dp = dst + (size_t)s * EMB + (lane16 << 2);
        atomic4_add_f32(dp, w * v.x, w * v.y, w * v.z, w * v.w);
      }
    }
    buf ^= 1;
  }
}

// ---------------------------------------------------------------------------
// Elementwise helpers (float4, fully coalesced).
// Input/output streams use NT hints so they don't evict the L2-resident
// embedding tables; table writes (buf) stay RT.
// ---------------------------------------------------------------------------

// outSum = in * LAYER_SCALE ; buf = in (buf optional)
__global__ void __launch_bounds__(256)
init_kernel(const v4f* __restrict__ in, v4f* __restrict__ outSum,
            v4f* __restrict__ buf, int n4) {
  int i = blockIdx.x * blockDim.x + threadIdx.x;
  if (i < n4) {
    v4f v = __builtin_nontemporal_load(in + i);
    __builtin_nontemporal_store(v * LAYER_SCALE, outSum + i);
    if (buf) buf[i] = v;   // table copy: keep RT / L2-resident
  }
}

__global__ void __launch_bounds__(256)
zero_kernel(v4f* __restrict__ p, int n4) {
  int i = blockIdx.x * blockDim.x + threadIdx.x;
  if (i < n4) {
    p[i] = (v4f)0.0f;      // table zero: keep RT / L2-resident
  }
}

// out += buf * LAYER_SCALE
__global__ void __launch_bounds__(256)
accum_kernel(v4f* __restrict__ out, const v4f* __restrict__ buf, int n4) {
  int i = blockIdx.x * blockDim.x + threadIdx.x;
  if (i < n4) {
    v4f o = __builtin_nontemporal_load(out + i);
    v4f b = buf[i];        // freshly built layer: already (and staying) in L2
    __builtin_nontemporal_store(o + b * LAYER_SCALE, out + i);
  }
}

// ---------------------------------------------------------------------------
// Host launcher
// ---------------------------------------------------------------------------
extern "C" void kernel_launch(void* const* d_in, const int* in_sizes, int n_in,
                              void* d_out, int out_size, void* d_ws, size_t ws_size,
                              hipStream_t stream) {
  const float* user_w  = (const float*)d_in[0];
  const float* item_w  = (const float*)d_in[1];
  const int*   u_idx   = (const int*)d_in[2];
  const int*   i_idx   = (const int*)d_in[3];
  const float* vals_ui = (const float*)d_in[4];
  const float* vals_iu = (const float*)d_in[5];

  const int nU = in_sizes[0] / EMB;   // 200000
  const int nI = in_sizes[1] / EMB;   // 100000
  const int nE = in_sizes[2];         // 2000000

  float* u_out = (float*)d_out;
  float* i_out = u_out + (size_t)nU * EMB;

  // workspace: u_buf (nU*EMB f32) then i_buf (nI*EMB f32) = 76.8 MB total
  float* u_buf = (float*)d_ws;
  float* i_buf = u_buf + (size_t)nU * EMB;

  const int nU4 = nU * EMB / 4;
  const int nI4 = nI * EMB / 4;
  const int gU4 = (nU4 + 255) / 256;
  const int gI4 = (nI4 + 255) / 256;
  const int gE  = (nE + EPB - 1) / EPB;
  const dim3 B(256);

  // u_sum = user_w/4, u_buf = user_w ; i_sum = item_w/4
  init_kernel<<<gU4, B, 0, stream>>>((const v4f*)user_w, (v4f*)u_out,
                                     (v4f*)u_buf, nU4);
  init_kernel<<<gI4, B, 0, stream>>>((const v4f*)item_w, (v4f*)i_out,
                                     (v4f*)nullptr, nI4);

  for (int l = 0; l < NLAYERS; ++l) {
    // i_{l+1} = M_iu @ u_l : gather u_idx, scatter i_idx, weight vals_iu
    zero_kernel<<<gI4, B, 0, stream>>>((v4f*)i_buf, nI4);
    scatter_spmm<<<gE, B, 0, stream>>>(i_buf, u_buf, u_idx, i_idx, vals_iu, nE);
    accum_kernel<<<gI4, B, 0, stream>>>((v4f*)i_out, (const v4f*)i_buf, nI4);

    // u_{l+1} = M_ui @ i_{l+1} : gather i_idx, scatter u_idx, weight vals_ui
    zero_kernel<<<gU4, B, 0, stream>>>((v4f*)u_buf, nU4);
    scatter_spmm<<<gE, B, 0, stream>>>(u_buf, i_buf, i_idx, u_idx, vals_ui, nE);
    accum_kernel<<<gU4, B, 0, stream>>>((v4f*)u_out, (const v4f*)u_buf, nU4);
  }
}